// SpatialProcessor_73864847557261
// MI455X (gfx1250) — compile-verified
//
#include <hip/hip_runtime.h>
#include <math.h>

typedef __attribute__((ext_vector_type(2))) float v2f;
typedef __attribute__((ext_vector_type(8))) float v8f;

#define NEG_SLOPE 0.2f
#define GAT_EPS 1e-9f

// ---------------------------------------------------------------- utilities

__global__ void fill_kernel(float* __restrict__ p, long n, float v) {
    long i = (long)blockIdx.x * blockDim.x + threadIdx.x;
    long stride = (long)gridDim.x * blockDim.x;
    for (; i < n; i += stride) p[i] = v;
}

__device__ __forceinline__ void atomicMaxFloat(float* addr, float val) {
    // IEEE monotone bit trick: ints compare like floats for >=0,
    // unsigned compares reversed for <0.
    if (val >= 0.0f) {
        atomicMax((int*)addr, __float_as_int(val));
    } else {
        atomicMin((unsigned int*)addr, __float_as_uint(val));
    }
}

// -------------------------------------------- WMMA f32 GEMM (16x16x4 tiles)
// C[M,N] = A[M,K] * B[K,N], all row-major f32.
// Requires M%16==0, N%64==0, K%4==0 (true here: M=50000=16*3125,
// N in {256,64}, K in {64,256}).
// One wave per 16x64 output strip: 4 independent accumulators reuse one A
// fragment per K-step (4 back-to-back WMMAs, no RAW hazard between them).
// Wave-uniform guard keeps EXEC all-ones at every WMMA.
__global__ void gemm_wmma_f32(const float* __restrict__ A,
                              const float* __restrict__ B,
                              float* __restrict__ C,
                              int M, int N, int K) {
    int wave = (int)((blockIdx.x * blockDim.x + threadIdx.x) >> 5);
    int lane = threadIdx.x & 31;
    int strips_n = N >> 6;                 // 64-wide strips
    int total = (M >> 4) * strips_n;
    if (wave >= total) return;             // uniform per wave
    int tm = wave / strips_n;
    int tg = wave % strips_n;

    int rl = lane & 15;                    // 0..15
    int hi = lane >> 4;                    // 0 (lanes 0-15) or 1 (lanes 16-31)

    const float* Arow  = A + (size_t)(tm * 16 + rl) * K;  // lane's A row (M=rl)
    const float* Bbase = B + (size_t)(tg * 64 + rl);      // lane's B column base

    v8f c0 = {}, c1 = {}, c2 = {}, c3 = {};
    for (int k0 = 0; k0 < K; k0 += 4) {
        int k = k0 + hi * 2;
        // A 16x4: VGPR0 = K=(0|2), VGPR1 = K=(1|3)  -> one b64 load
        v2f a;
        a.x = Arow[k];
        a.y = Arow[k + 1];
        // B 4x16 fragments for 4 adjacent N-tiles (cols +0,+16,+32,+48)
        const float* Br0 = Bbase + (size_t)k * N;       // row K=(0|2)
        const float* Br1 = Br0 + N;                     // row K=(1|3)
        v2f b0, b1, b2, b3;
        b0.x = Br0[0];  b0.y = Br1[0];
        b1.x = Br0[16]; b1.y = Br1[16];
        b2.x = Br0[32]; b2.y = Br1[32];
        b3.x = Br0[48]; b3.y = Br1[48];
        // 8 args: (neg_a, A, neg_b, B, c_mod, C, reuse_a, reuse_b)
        c0 = __builtin_amdgcn_wmma_f32_16x16x4_f32(false, a, false, b0,
                                                   (short)0, c0, false, false);
        c1 = __builtin_amdgcn_wmma_f32_16x16x4_f32(false, a, false, b1,
                                                   (short)0, c1, false, false);
        c2 = __builtin_amdgcn_wmma_f32_16x16x4_f32(false, a, false, b2,
                                                   (short)0, c2, false, false);
        c3 = __builtin_amdgcn_wmma_f32_16x16x4_f32(false, a, false, b3,
                                                   (short)0, c3, false, false);
    }

    // C/D layout: VGPR v -> row M = v + (hi ? 8 : 0), col = lane&15
    float* Crow = C + (size_t)(tm * 16 + hi * 8) * N + tg * 64 + rl;
#pragma unroll
    for (int v = 0; v < 8; ++v) {
        Crow[(size_t)v * N + 0]  = c0[v];
        Crow[(size_t)v * N + 16] = c1[v];
        Crow[(size_t)v * N + 32] = c2[v];
        Crow[(size_t)v * N + 48] = c3[v];
    }
}

// -------------------------------------------- per-(node,head) score vectors
// ss[n,h] = sum_d h[n,h,d]*a_src[h,d];  sd likewise with a_dst.
__global__ void scores_kernel(const float* __restrict__ hfeat,
                              const float* __restrict__ a_src,
                              const float* __restrict__ a_dst,
                              float* __restrict__ ss, float* __restrict__ sd,
                              int N, int H, int D) {
    int i = blockIdx.x * blockDim.x + threadIdx.x;    // i = n*H + h
    if (i >= N * H) return;
    int head = i % H;
    const float* hv = hfeat + (size_t)i * D;
    const float* as = a_src + (size_t)head * D;
    const float* ad = a_dst + (size_t)head * D;
    float accs = 0.0f, accd = 0.0f;
    for (int d = 0; d < D; ++d) {
        float v = hv[d];
        accs += v * as[d];
        accd += v * ad[d];
    }
    ss[i] = accs;
    sd[i] = accd;
}

// ------------------------------------------------ edge pass 1: segment max
__global__ void edge_max_kernel(const int2* __restrict__ edges,
                                const float* __restrict__ ss,
                                const float* __restrict__ sd,
                                float* __restrict__ emax, int E, int H) {
    int e = blockIdx.x * blockDim.x + threadIdx.x;
    if (e >= E) return;
    int2 ed = edges[e];                    // .x = dst, .y = src (b64 load)
    int dst = ed.x, src = ed.y;
    for (int h = 0; h < H; ++h) {
        float v = ss[src * H + h] + sd[dst * H + h];
        v = (v > 0.0f) ? v : NEG_SLOPE * v;           // leaky relu
        atomicMaxFloat(&emax[dst * H + h], v);
    }
}

__global__ void fix_emax_kernel(float* __restrict__ emax, int n) {
    int i = blockIdx.x * blockDim.x + threadIdx.x;
    if (i >= n) return;
    float v = emax[i];
    if (!__builtin_isfinite(v)) emax[i] = 0.0f;
}

// --------------------- edge pass 2: exp, denom, unnormalized msg accumulate
// One wave per edge: scalar ex per head, 32 lanes stream the D features.
// alpha = ex/denom is deferred: accumulate sum(ex*h[src]) and denom, divide
// once per node afterwards (exactly matches reference algebra).
__global__ void edge_accum_kernel(const int2* __restrict__ edges,
                                  const float* __restrict__ ss,
                                  const float* __restrict__ sd,
                                  const float* __restrict__ emax,
                                  const float* __restrict__ hfeat,
                                  float* __restrict__ denom,
                                  float* __restrict__ acc,
                                  int E, int H, int D) {
    int wave = (int)((blockIdx.x * blockDim.x + threadIdx.x) >> 5);
    int lane = threadIdx.x & 31;
    if (wave >= E) return;                 // uniform per wave
    int2 ed = edges[wave];
    int dst = ed.x, src = ed.y;
    for (int h = 0; h < H; ++h) {
        float v = ss[src * H + h] + sd[dst * H + h];
        v = (v > 0.0f) ? v : NEG_SLOPE * v;
        float ex = __expf(v - emax[dst * H + h]);
        if (lane == 0) atomicAdd(&denom[dst * H + h], ex);
        const float* hs = hfeat + ((size_t)src * H + h) * D;
        float* ad = acc + ((size_t)dst * H + h) * D;
        for (int d = lane; d < D; d += 32) {
            atomicAdd(&ad[d], ex * hs[d]);
        }
    }
}

// --------------------------- normalize by denom, add bias, optional ReLU
__global__ void normalize_kernel(const float* __restrict__ acc,
                                 const float* __restrict__ denom,
                                 const float* __restrict__ bias,
                                 float* __restrict__ out,
                                 int N, int H, int D, int do_relu) {
    long total = (long)N * H * D;
    long i = (long)blockIdx.x * blockDim.x + threadIdx.x;
    if (i >= total) return;
    int hd = (int)(i % ((long)H * D));     // bias index (concat-heads layout)
    long nh = i / D;                       // n*H + h
    float v = acc[i] / (denom[nh] + GAT_EPS) + bias[hd];
    if (do_relu) v = fmaxf(v, 0.0f);
    out[i] = v;
}

// ---------------------------------------------------------------- launcher

extern "C" void kernel_launch(void* const* d_in, const int* in_sizes, int n_in,
                              void* d_out, int out_size, void* d_ws, size_t ws_size,
                              hipStream_t stream) {
    const float* x      = (const float*)d_in[0];   // [N,64]
    const int2*  edges  = (const int2*) d_in[1];   // [E,2] (dst, src)
    const float* W1     = (const float*)d_in[2];   // [64,4,64] == [64][256]
    const float* a_src1 = (const float*)d_in[3];   // [4,64]
    const float* a_dst1 = (const float*)d_in[4];
    const float* b1     = (const float*)d_in[5];   // [256]
    const float* W2     = (const float*)d_in[6];   // [256,1,64] == [256][64]
    const float* a_src2 = (const float*)d_in[7];   // [1,64]
    const float* a_dst2 = (const float*)d_in[8];
    const float* b2     = (const float*)d_in[9];   // [64]
    float* out = (float*)d_out;                    // [N,64]

    const int N = in_sizes[0] / 64;                // 50000
    const int E = in_sizes[1] / 2;                 // 800800
    const int Fin = 64, H1 = 4, D = 64, F1 = H1 * D;   // 256

    // ---- workspace layout (floats) ----
    float* h1   = (float*)d_ws;                    // N*256 (layer1 feats; reused as h2)
    float* acc1 = h1 + (size_t)N * F1;             // N*256 (numerator; then layer2 input)
    float* ss1  = acc1 + (size_t)N * F1;           // N*4
    float* sd1  = ss1 + (size_t)N * H1;            // N*4
    float* em1  = sd1 + (size_t)N * H1;            // N*4
    float* dn1  = em1 + (size_t)N * H1;            // N*4
    float* ss2  = dn1 + (size_t)N * H1;            // N
    float* sd2  = ss2 + (size_t)N;                 // N
    float* em2  = sd2 + (size_t)N;                 // N
    float* dn2  = em2 + (size_t)N;                 // N

    const int TB = 256;
    const int FILLB = 2048;

    // ================= Layer 1 =================
    fill_kernel<<<FILLB, TB, 0, stream>>>(acc1, (long)N * F1, 0.0f);
    fill_kernel<<<FILLB, TB, 0, stream>>>(dn1, (long)N * H1, 0.0f);
    fill_kernel<<<FILLB, TB, 0, stream>>>(em1, (long)N * H1, -INFINITY);

    {   // h1 = x @ W1   ([N,64]x[64,256]); 16x64 strip per wave
        int waves = (N / 16) * (F1 / 64);
        int blocks = (int)(((long)waves * 32 + TB - 1) / TB);
        gemm_wmma_f32<<<blocks, TB, 0, stream>>>(x, W1, h1, N, F1, Fin);
    }
    scores_kernel<<<(N * H1 + TB - 1) / TB, TB, 0, stream>>>(h1, a_src1, a_dst1,
                                                             ss1, sd1, N, H1, D);
    edge_max_kernel<<<(E + TB - 1) / TB, TB, 0, stream>>>(edges, ss1, sd1, em1, E, H1);
    fix_emax_kernel<<<(N * H1 + TB - 1) / TB, TB, 0, stream>>>(em1, N * H1);
    {   // wave per edge
        int blocks = (int)(((long)E * 32 + TB - 1) / TB);
        edge_accum_kernel<<<blocks, TB, 0, stream>>>(edges, ss1, sd1, em1, h1,
                                                     dn1, acc1, E, H1, D);
    }
    {   // acc1 <- relu(acc1/denom + b1)  (in place; becomes layer-2 input)
        long total = (long)N * F1;
        normalize_kernel<<<(int)((total + TB - 1) / TB), TB, 0, stream>>>(
            acc1, dn1, b1, acc1, N, H1, D, 1);
    }

    // ================= Layer 2 =================
    fill_kernel<<<FILLB, TB, 0, stream>>>(out, (long)N * D, 0.0f);
    fill_kernel<<<FILLB, TB, 0, stream>>>(dn2, (long)N, 0.0f);
    fill_kernel<<<FILLB, TB, 0, stream>>>(em2, (long)N, -INFINITY);

    {   // h2 = acc1 @ W2   ([N,256]x[256,64]) -> reuse h1 buffer
        int waves = (N / 16) * (D / 64);
        int blocks = (int)(((long)waves * 32 + TB - 1) / TB);
        gemm_wmma_f32<<<blocks, TB, 0, stream>>>(acc1, W2, h1, N, D, F1);
    }
    scores_kernel<<<(N + TB - 1) / TB, TB, 0, stream>>>(h1, a_src2, a_dst2,
                                                        ss2, sd2, N, 1, D);
    edge_max_kernel<<<(E + TB - 1) / TB, TB, 0, stream>>>(edges, ss2, sd2, em2, E, 1);
    fix_emax_kernel<<<(N + TB - 1) / TB, TB, 0, stream>>>(em2, N);
    {
        int blocks = (int)(((long)E * 32 + TB - 1) / TB);
        edge_accum_kernel<<<blocks, TB, 0, stream>>>(edges, ss2, sd2, em2, h1,
                                                     dn2, out, E, 1, D);
    }
    {   // out <- out/denom + b2  (H=1 so mean over heads is identity)
        long total = (long)N * D;
        normalize_kernel<<<(int)((total + TB - 1) / TB), TB, 0, stream>>>(
            out, dn2, b2, out, N, 1, D, 0);
    }
}